// Net_18829136626136
// MI455X (gfx1250) — compile-verified
//
#include <hip/hip_runtime.h>
#include <math.h>

// ============================================================================
// PointTransformer-style U-Net forward on gfx1250 (MI455X).
// - All dense matmuls through V_WMMA_F32_16X16X4_F32 (fp32 in/out, matches
//   the f32 reference numerics; WMMA is CDNA5's only matrix path).
// - GEMM: LDS staging with transposed, stride-20-padded tiles so every A/B
//   fragment is one bank-conflict-free ds_load_b64 straight into the
//   even-aligned VGPR pair WMMA wants; inner loop is branch-free.
//   Staging has a b128 fast path for fully in-bounds tiles and a clamp+select
//   (branch-free, no EXEC gymnastics) path for edge tiles.
// - Irregular parts (kNN, FPS, softmax aggregation) as VALU kernels.
// - Entire live working set (~50MB) is L2-resident (192MB L2), so the pass
//   is launch/latency bound; fp32 WMMA throughput is not the bottleneck.
// ============================================================================

typedef float v2f __attribute__((ext_vector_type(2)));
typedef float v8f __attribute__((ext_vector_type(8)));

// ---------------------------------------------------------------------------
// GEMM: C[N,M] = A[N,K] @ W[K,M] (+bias) (+relu), row-major everything.
// Block = 128 threads = 4 waves. Block tile 64x64; each wave owns a 16x64
// strip (4 WMMA col-tiles, A fragment reused across them). K consumed in
// chunks of 16 staged through LDS (zero-padded at edges): 16 v_wmma/chunk.
//   flags bit0 = add bias, bit1 = relu
// LDS: As[row][k], Bs[col][k] (transposed), both stride 20 floats:
//  - fragment = 2 contiguous floats at even offset -> single ds_load_b64
//  - r*20 mod 64 distinct for r=0..15 and 20d == -2 (mod 64) unsolvable,
//    so fragment loads are bank-conflict-free across the full wave.
// Fragment layouts per ISA 7.12.2 (32-bit):
//   A 16x4: lane<16 row=lane K={0,1}; lane>=16 K={2,3}.
//   B 4x16: mirrored (col=lane&15, K pair selected by lane>=16).
//   C/D 16x16: VGPR j -> row j (lanes 0-15) / j+8 (lanes 16-31), col=lane&15.
// ---------------------------------------------------------------------------
#define GEMM_KC 16
#define LDS_STR 20

__global__ void gemm_f32_wmma(const float* __restrict__ A,
                              const float* __restrict__ W,
                              const float* __restrict__ bias,
                              float* __restrict__ C,
                              int N, int K, int M, int flags)
{
  const int lane = threadIdx.x & 31;
  const int wave = threadIdx.x >> 5;
  const int rowBlock = blockIdx.y * 64;
  const int colBase  = blockIdx.x * 64;
  const int rowBase  = rowBlock + wave * 16;
  const int r  = lane & 15;
  const int kh = (lane >> 4) << 1;    // 0 or 2

  __shared__ float As[64 * LDS_STR];  // [row][k], stride 20
  __shared__ float Bs[64 * LDS_STR];  // [col][k], stride 20 (transposed)

  v8f zero = {0.f,0.f,0.f,0.f,0.f,0.f,0.f,0.f};
  v8f acc[4];
  acc[0] = zero; acc[1] = zero; acc[2] = zero; acc[3] = zero;

  // Block-uniform: interior tile with vectorizable strides -> b128 staging.
  const bool fullTile = (rowBlock + 64 <= N) && (colBase + 64 <= M) &&
                        ((K & 3) == 0) && ((M & 3) == 0);

  for (int k0 = 0; k0 < K; k0 += GEMM_KC) {
    __syncthreads();
    if (fullTile && (k0 + GEMM_KC <= K)) {
      // ---- fast path: unguarded 128-bit loads ----
      // A: 64 rows x 16 k = 256 float4, 2 per thread; ds_store_b128
      // (LDS offset rr*20 + k4 with k4 in {0,4,8,12} -> 16B aligned).
#pragma unroll
      for (int u = 0; u < 2; ++u) {
        const int i4 = (int)threadIdx.x + u * 128;
        const int rr = i4 >> 2;
        const int k4 = (i4 & 3) << 2;
        const float4 v = *(const float4*)&A[(size_t)(rowBlock + rr) * K + k0 + k4];
        *(float4*)&As[rr * LDS_STR + k4] = v;
      }
      // B: 16 k x 64 cols = 256 float4 along cols; transposed scatter.
#pragma unroll
      for (int u = 0; u < 2; ++u) {
        const int i4 = (int)threadIdx.x + u * 128;
        const int kk = i4 >> 4;
        const int c4 = (i4 & 15) << 2;
        const float4 v = *(const float4*)&W[(size_t)(k0 + kk) * M + colBase + c4];
        Bs[(c4 + 0) * LDS_STR + kk] = v.x;
        Bs[(c4 + 1) * LDS_STR + kk] = v.y;
        Bs[(c4 + 2) * LDS_STR + kk] = v.z;
        Bs[(c4 + 3) * LDS_STR + kk] = v.w;
      }
    } else {
      // ---- edge path: branch-free clamp + select (no EXEC masking) ----
#pragma unroll
      for (int u = 0; u < 8; ++u) {
        const int idx = (int)threadIdx.x + u * 128;
        const int rr = idx >> 4;
        const int kk = idx & (GEMM_KC - 1);
        const int gr = rowBlock + rr;
        const int gk = k0 + kk;
        const bool in = (gr < N) && (gk < K);
        const size_t o = in ? ((size_t)gr * K + gk) : 0;
        const float v = A[o];
        As[rr * LDS_STR + kk] = in ? v : 0.0f;
      }
#pragma unroll
      for (int u = 0; u < 8; ++u) {
        const int idx = (int)threadIdx.x + u * 128;
        const int kk = idx >> 6;
        const int cc = idx & 63;
        const int gk = k0 + kk;
        const int gc = colBase + cc;
        const bool in = (gk < K) && (gc < M);
        const size_t o = in ? ((size_t)gk * M + gc) : 0;
        const float v = W[o];
        Bs[cc * LDS_STR + kk] = in ? v : 0.0f;
      }
    }
    __syncthreads();

    // Branch-free WMMA loop: 4 k-steps x 4 col-tiles = 16 v_wmma per chunk.
    const float* Arow = &As[(wave * 16 + r) * LDS_STR];
#pragma unroll
    for (int ks = 0; ks < GEMM_KC; ks += 4) {
      const v2f a = *(const v2f*)&Arow[ks + kh];          // ds_load_b64
#pragma unroll
      for (int t = 0; t < 4; ++t) {
        const v2f b = *(const v2f*)&Bs[(t * 16 + r) * LDS_STR + ks + kh];
        acc[t] = __builtin_amdgcn_wmma_f32_16x16x4_f32(
            false, a, false, b, (short)0, acc[t], false, false);
      }
    }
  }

  const int rext = (lane >> 4) << 3;  // +8 for high half-wave
#pragma unroll
  for (int t = 0; t < 4; ++t) {
    const int col = colBase + t * 16 + r;
    if (col >= M) continue;
    const float bv = (flags & 1) ? bias[col] : 0.0f;
#pragma unroll
    for (int j = 0; j < 8; ++j) {
      const int row = rowBase + j + rext;
      if (row >= N) continue;
      float v = acc[t][j] + bv;
      if (flags & 2) v = fmaxf(v, 0.0f);
      C[(size_t)row * M + col] = v;
    }
  }
}

// ---------------------------------------------------------------------------
// Per-column mean/var over N rows (training-mode BN stats, biased variance).
// One block per column.
// ---------------------------------------------------------------------------
__global__ void colstats_kernel(const float* __restrict__ X, int N, int C,
                                float* __restrict__ mean, float* __restrict__ var)
{
  const int c = blockIdx.x;
  float s = 0.f, s2 = 0.f;
  for (int i = threadIdx.x; i < N; i += blockDim.x) {
    float v = X[(size_t)i * C + c];
    s += v; s2 += v * v;
  }
  __shared__ float sh[256], sh2[256];
  sh[threadIdx.x] = s; sh2[threadIdx.x] = s2;
  __syncthreads();
  for (int o = blockDim.x >> 1; o > 0; o >>= 1) {
    if ((int)threadIdx.x < o) {
      sh[threadIdx.x]  += sh[threadIdx.x + o];
      sh2[threadIdx.x] += sh2[threadIdx.x + o];
    }
    __syncthreads();
  }
  if (threadIdx.x == 0) {
    float m = sh[0] / (float)N;
    mean[c] = m;
    var[c]  = sh2[0] / (float)N - m * m;
  }
}

__global__ void bn_relu_kernel(const float* __restrict__ x, float* __restrict__ y,
                               int N, int C,
                               const float* __restrict__ mean,
                               const float* __restrict__ var)
{
  int t = blockIdx.x * blockDim.x + threadIdx.x;
  if (t >= N * C) return;
  int c = t % C;
  float v = (x[t] - mean[c]) * rsqrtf(var[c] + 1e-5f);
  y[t] = fmaxf(v, 0.0f);
}

// ---------------------------------------------------------------------------
// Brute-force kNN: for each query, top-k smallest squared distances over base.
// One thread per query; base points staged through LDS in tiles of 256.
// exclude_self==1 skips j == global query index (knn_graph, loop=False).
// Insertion with strict '>' keeps the lower index on ties (top_k-stable).
// ---------------------------------------------------------------------------
#define KNN_MAXK 16
__global__ void knn_kernel(const float* __restrict__ qpos,
                           const float* __restrict__ bpos,
                           int Nq, int Nb, int k, int exclude_self,
                           int* __restrict__ outIdx)
{
  const int q = blockIdx.x * blockDim.x + threadIdx.x;
  float qx = 0.f, qy = 0.f, qz = 0.f;
  if (q < Nq) { qx = qpos[q*3+0]; qy = qpos[q*3+1]; qz = qpos[q*3+2]; }

  float bestD[KNN_MAXK];
  int   bestI[KNN_MAXK];
  for (int j = 0; j < KNN_MAXK; ++j) { bestD[j] = INFINITY; bestI[j] = 0; }

  __shared__ float sb[256 * 3];
  for (int base = 0; base < Nb; base += 256) {
    const int nb = (Nb - base < 256) ? (Nb - base) : 256;
    __syncthreads();
    for (int t = threadIdx.x; t < nb * 3; t += blockDim.x)
      sb[t] = bpos[(size_t)base * 3 + t];
    __syncthreads();
    if (q < Nq) {
      for (int j = 0; j < nb; ++j) {
        const int gj = base + j;
        if (exclude_self && gj == q) continue;
        float dx = qx - sb[j*3+0];
        float dy = qy - sb[j*3+1];
        float dz = qz - sb[j*3+2];
        float d = dx*dx + dy*dy + dz*dz;
        if (d < bestD[k - 1]) {
          int p = k - 1;
          while (p > 0 && bestD[p-1] > d) {
            bestD[p] = bestD[p-1]; bestI[p] = bestI[p-1]; --p;
          }
          bestD[p] = d; bestI[p] = gj;
        }
      }
    }
  }
  if (q < Nq)
    for (int j = 0; j < k; ++j) outIdx[(size_t)q * k + j] = bestI[j];
}

// ---------------------------------------------------------------------------
// Farthest point sampling, single block, deterministic start at index 0.
// Matches reference: dists=inf, idxs[0]=0, argmax ties -> lowest index.
// ---------------------------------------------------------------------------
__global__ void fps_kernel(const float* __restrict__ pos, int N, int n,
                           int* __restrict__ outIdx, float* __restrict__ dists)
{
  __shared__ float smax[256];
  __shared__ int   simax[256];
  __shared__ int   s_last;
  if (threadIdx.x == 0) { outIdx[0] = 0; s_last = 0; }
  for (int i = threadIdx.x; i < N; i += blockDim.x) dists[i] = INFINITY;
  __syncthreads();

  for (int it = 1; it < n; ++it) {
    const int last = s_last;
    const float lx = pos[last*3+0], ly = pos[last*3+1], lz = pos[last*3+2];
    float bm = -1.0f; int bi = N;
    for (int i = threadIdx.x; i < N; i += blockDim.x) {
      float dx = pos[i*3+0] - lx;
      float dy = pos[i*3+1] - ly;
      float dz = pos[i*3+2] - lz;
      float d = dx*dx + dy*dy + dz*dz;
      float cur = dists[i];
      if (d < cur) { cur = d; dists[i] = d; }
      if (cur > bm) { bm = cur; bi = i; }       // strict > keeps lowest i
    }
    smax[threadIdx.x] = bm; simax[threadIdx.x] = bi;
    __syncthreads();
    for (int o = blockDim.x >> 1; o > 0; o >>= 1) {
      if ((int)threadIdx.x < o) {
        float ov = smax[threadIdx.x + o]; int oi = simax[threadIdx.x + o];
        if (ov > smax[threadIdx.x] ||
            (ov == smax[threadIdx.x] && oi < simax[threadIdx.x])) {
          smax[threadIdx.x] = ov; simax[threadIdx.x] = oi;
        }
      }
      __syncthreads();
    }
    if (threadIdx.x == 0) { outIdx[it] = simax[0]; s_last = simax[0]; }
    __syncthreads();
  }
}

// ---------------------------------------------------------------------------
__global__ void gather_rows_kernel(const float* __restrict__ src,
                                   const int* __restrict__ idx,
                                   float* __restrict__ out, int n, int cols)
{
  int t = blockIdx.x * blockDim.x + threadIdx.x;
  if (t >= n * cols) return;
  int i = t / cols, c = t - i * cols;
  out[t] = src[(size_t)idx[i] * cols + c];
}

__global__ void gather_max_kernel(const float* __restrict__ x,
                                  const int* __restrict__ nb,
                                  float* __restrict__ out, int n, int k, int C)
{
  int t = blockIdx.x * blockDim.x + threadIdx.x;
  if (t >= n * C) return;
  int i = t / C, c = t - i * C;
  float m = -INFINITY;
  for (int j = 0; j < k; ++j)
    m = fmaxf(m, x[(size_t)nb[(size_t)i * k + j] * C + c]);
  out[t] = m;
}

// Edge list layout (matches reference concatenation order):
//   e in [0, N*k):  src = nbr[e], dst = e / k
//   e in [N*k, N*k+N): self loop, src = dst = e - N*k
__global__ void edge_posdiff_kernel(const float* __restrict__ pos,
                                    const int* __restrict__ nbr,
                                    float* __restrict__ pd, int N, int k)
{
  int e = blockIdx.x * blockDim.x + threadIdx.x;
  int E = N * (k + 1);
  if (e >= E) return;
  int src, dst;
  if (e < N * k) { src = nbr[e]; dst = e / k; }
  else           { dst = e - N * k; src = dst; }
  pd[(size_t)e*3+0] = pos[dst*3+0] - pos[src*3+0];
  pd[(size_t)e*3+1] = pos[dst*3+1] - pos[src*3+1];
  pd[(size_t)e*3+2] = pos[dst*3+2] - pos[src*3+2];
}

__global__ void edge_attnin_kernel(const float* __restrict__ AD,
                                   const float* __restrict__ AS,
                                   const float* __restrict__ DELTA,
                                   const int* __restrict__ nbr,
                                   float* __restrict__ out, int N, int k, int C)
{
  int t = blockIdx.x * blockDim.x + threadIdx.x;
  int E = N * (k + 1);
  if (t >= E * C) return;
  int e = t / C, c = t - e * C;
  int src, dst;
  if (e < N * k) { src = nbr[e]; dst = e / k; }
  else           { dst = e - N * k; src = dst; }
  out[t] = AD[(size_t)dst * C + c] - AS[(size_t)src * C + c] + DELTA[t];
}

// Per-target, per-channel softmax over the (k+1) edges of node i, then
// weighted sum of (v[src] + delta). Matches reference segment ops exactly.
__global__ void pt_aggregate_kernel(const float* __restrict__ alpha,
                                    const float* __restrict__ delta,
                                    const float* __restrict__ v,
                                    const int* __restrict__ nbr,
                                    float* __restrict__ out, int N, int k, int C)
{
  int t = blockIdx.x * blockDim.x + threadIdx.x;
  if (t >= N * C) return;
  int i = t / C, c = t - i * C;
  const int kk = k + 1;
  float av[KNN_MAXK + 1];
  float m = -INFINITY;
  for (int j = 0; j < kk; ++j) {
    int e = (j < k) ? (i * k + j) : (N * k + i);
    float a = alpha[(size_t)e * C + c];
    av[j] = a;
    m = fmaxf(m, a);
  }
  float s = 0.f;
  for (int j = 0; j < kk; ++j) { av[j] = expf(av[j] - m); s += av[j]; }
  const float inv = 1.0f / (s + 1e-16f);
  float acc = 0.f;
  for (int j = 0; j < kk; ++j) {
    int e, sidx;
    if (j < k) { e = i * k + j; sidx = nbr[e]; }
    else       { e = N * k + i; sidx = i; }
    acc += (av[j] * inv) *
           (v[(size_t)sidx * C + c] + delta[(size_t)e * C + c]);
  }
  out[t] = acc;
}

// Inverse-distance^2 interpolation from 3 nearest coarse points.
__global__ void interp3_kernel(const float* __restrict__ xsub,
                               const int* __restrict__ idx,
                               const float* __restrict__ pos,
                               const float* __restrict__ possub,
                               float* __restrict__ xint, int N, int C)
{
  int t = blockIdx.x * blockDim.x + threadIdx.x;
  if (t >= N * C) return;
  int i = t / C, c = t - i * C;
  float px = pos[i*3+0], py = pos[i*3+1], pz = pos[i*3+2];
  float wsum = 0.f, acc = 0.f;
  for (int j = 0; j < 3; ++j) {
    int s = idx[(size_t)i * 3 + j];
    float dx = possub[s*3+0] - px;
    float dy = possub[s*3+1] - py;
    float dz = possub[s*3+2] - pz;
    float d2 = dx*dx + dy*dy + dz*dz;
    float w = 1.0f / fmaxf(d2, 1e-16f);
    wsum += w;
    acc += xsub[(size_t)s * C + c] * w;
  }
  xint[t] = acc / wsum;
}

__global__ void add_inplace_kernel(float* __restrict__ a,
                                   const float* __restrict__ b, int n)
{
  int t = blockIdx.x * blockDim.x + threadIdx.x;
  if (t < n) a[t] += b[t];
}

__global__ void log_softmax_kernel(const float* __restrict__ h,
                                   float* __restrict__ out, int N, int C)
{
  int i = blockIdx.x * blockDim.x + threadIdx.x;
  if (i >= N) return;
  float m = -INFINITY;
  for (int c = 0; c < C; ++c) m = fmaxf(m, h[(size_t)i * C + c]);
  float s = 0.f;
  for (int c = 0; c < C; ++c) s += expf(h[(size_t)i * C + c] - m);
  float ls = logf(s) + m;
  for (int c = 0; c < C; ++c)
    out[(size_t)i * C + c] = h[(size_t)i * C + c] - ls;
}

// ============================================================================
// Host-side orchestration
// ============================================================================
namespace {

struct Lin { const float* W; const float* b; int din, dout; };
struct TBp { Lin attn0, attn1, lin, lin_dst, lin_in, lin_out, lin_src, pos0, pos1; };

struct Ws {
  char* base; size_t off;
  float* f(size_t n) { size_t a = (off + 255) & ~(size_t)255; off = a + n * 4; return (float*)(base + a); }
  int*   ip(size_t n){ size_t a = (off + 255) & ~(size_t)255; off = a + n * 4; return (int*)(base + a); }
};

inline unsigned blocks_for(size_t n) { return (unsigned)((n + 255) / 256); }

inline void gemmL(hipStream_t s, const float* A, const Lin& L, float* C, int N, bool relu) {
  int flags = (L.b ? 1 : 0) | (relu ? 2 : 0);
  dim3 grid((L.dout + 63) / 64, (N + 63) / 64);
  gemm_f32_wmma<<<grid, dim3(128), 0, s>>>(A, L.W, L.b, C, N, L.din, L.dout, flags);
}

inline void mlp_bn_run(hipStream_t s, Ws& ws, const float* A, const Lin& L,
                       float* out, int N) {
  size_t mark = ws.off;
  float* tmp  = ws.f((size_t)N * L.dout);
  float* mean = ws.f(L.dout);
  float* var  = ws.f(L.dout);
  gemmL(s, A, L, tmp, N, false);
  colstats_kernel<<<dim3(L.dout), dim3(256), 0, s>>>(tmp, N, L.dout, mean, var);
  bn_relu_kernel<<<blocks_for((size_t)N * L.dout), 256, 0, s>>>(tmp, out, N, L.dout, mean, var);
  ws.off = mark;
}

void tblock_run(hipStream_t s, Ws& ws, const TBp& tb, float* x,
                const float* pos, int N, int C) {
  size_t mark = ws.off;
  const int K = 16;
  const int E = N * (K + 1);

  int* nbr = ws.ip((size_t)N * K);
  knn_kernel<<<blocks_for(N), 256, 0, s>>>(pos, pos, N, N, K, 1, nbr);

  float* XH = ws.f((size_t)N * C);
  gemmL(s, x, tb.lin_in, XH, N, true);                       // relu(lin_in(x))

  float* AS = ws.f((size_t)N * C);
  float* AD = ws.f((size_t)N * C);
  float* V  = ws.f((size_t)N * C);
  gemmL(s, XH, tb.lin_src, AS, N, false);
  gemmL(s, XH, tb.lin_dst, AD, N, false);
  gemmL(s, XH, tb.lin,     V,  N, false);

  float* PD = ws.f((size_t)E * 3);
  edge_posdiff_kernel<<<blocks_for(E), 256, 0, s>>>(pos, nbr, PD, N, K);

  // Edge MLPs, chunked so the 64-wide hidden buffer stays small (L2-friendly).
  const int EC = 16384;
  float* DELTA = ws.f((size_t)E * C);
  const int hsz = (E < EC) ? E : EC;
  float* H = ws.f((size_t)hsz * 64);
  for (int o = 0; o < E; o += EC) {
    int n = (E - o < EC) ? (E - o) : EC;
    gemmL(s, PD + (size_t)o * 3, tb.pos0, H, n, true);               // 3 -> 64
    gemmL(s, H, tb.pos1, DELTA + (size_t)o * C, n, true);            // 64 -> C
  }

  float* AT = ws.f((size_t)E * C);
  edge_attnin_kernel<<<blocks_for((size_t)E * C), 256, 0, s>>>(AD, AS, DELTA, nbr, AT, N, K, C);
  for (int o = 0; o < E; o += EC) {
    int n = (E - o < EC) ? (E - o) : EC;
    gemmL(s, AT + (size_t)o * C, tb.attn0, H, n, true);              // C -> 64
    gemmL(s, H, tb.attn1, AT + (size_t)o * C, n, true);              // 64 -> C (alpha)
  }

  float* AGG = ws.f((size_t)N * C);
  pt_aggregate_kernel<<<blocks_for((size_t)N * C), 256, 0, s>>>(AT, DELTA, V, nbr, AGG, N, K, C);
  gemmL(s, AGG, tb.lin_out, x, N, true);                     // relu(lin_out(.)) -> x
  ws.off = mark;
}

} // namespace

extern "C" void kernel_launch(void* const* d_in, const int* in_sizes, int n_in,
                              void* d_out, int out_size, void* d_ws, size_t ws_size,
                              hipStream_t stream)
{
  (void)in_sizes; (void)n_in; (void)out_size; (void)ws_size;
  const float* x_in = (const float*)d_in[0];   // [8192, 6]
  const float* pos0 = (const float*)d_in[1];   // [8192, 3]

  // --- parameter unpacking: jax pytree leaf order (dict keys sorted) ---
  int p = 2;
  auto nextLin = [&](int din, int dout, bool bias) -> Lin {
    Lin l; l.W = (const float*)d_in[p++];
    l.b = bias ? (const float*)d_in[p++] : nullptr;
    l.din = din; l.dout = dout; return l;
  };
  auto nextTB = [&](int c) -> TBp {
    TBp t;                                 // keys sorted alphabetically:
    t.attn0   = nextLin(c, 64, true);      // attn_nn[0]
    t.attn1   = nextLin(64, c, true);      // attn_nn[1]
    t.lin     = nextLin(c, c, false);      // lin
    t.lin_dst = nextLin(c, c, false);      // lin_dst
    t.lin_in  = nextLin(c, c, true);       // lin_in
    t.lin_out = nextLin(c, c, true);       // lin_out
    t.lin_src = nextLin(c, c, false);      // lin_src
    t.pos0    = nextLin(3, 64, true);      // pos_nn[0]
    t.pos1    = nextLin(64, c, true);      // pos_nn[1]
    return t;
  };

  static const int DIMS[5] = {32, 64, 128, 256, 512};
  static const int Ns[5]   = {8192, 2048, 512, 128, 32};

  // top-level sorted keys: dec, enc, mlp_input, out, summit, tb_input, tb_summit
  Lin dec_mlp[4], dec_sub[4]; TBp dec_tb[4];
  for (int i = 0; i < 4; ++i) {            // dec[i] keys: mlp, mlp_sub, tb
    int Cl = DIMS[3 - i], Ch = DIMS[4 - i];
    dec_mlp[i] = nextLin(Cl, Cl, true);
    dec_sub[i] = nextLin(Ch, Cl, true);
    dec_tb[i]  = nextTB(Cl);
  }
  TBp enc_tb[4]; Lin enc_td[4];
  for (int i = 0; i < 4; ++i) {            // enc[i] keys: tb, td_mlp
    enc_tb[i] = nextTB(DIMS[i + 1]);
    enc_td[i] = nextLin(DIMS[i], DIMS[i + 1], true);
  }
  Lin mlp_input = nextLin(6, 32, true);
  Lin out0 = nextLin(32, 64, true);
  Lin out1 = nextLin(64, 64, true);
  Lin out2 = nextLin(64, 13, true);
  Lin summit = nextLin(512, 512, true);
  TBp tb_input  = nextTB(32);
  TBp tb_summit = nextTB(512);

  // --- workspace ---
  Ws ws; ws.base = (char*)d_ws; ws.off = 0;
  float* X[5];
  const float* POS[5];
  float* POSb[5] = {nullptr, nullptr, nullptr, nullptr, nullptr};
  X[0] = ws.f((size_t)Ns[0] * DIMS[0]);
  POS[0] = pos0;
  for (int l = 1; l < 5; ++l) {
    X[l]   = ws.f((size_t)Ns[l] * DIMS[l]);
    POSb[l]= ws.f((size_t)Ns[l] * 3);
    POS[l] = POSb[l];
  }
  float* XS = ws.f((size_t)Ns[4] * 512);   // summit features

  // --- level 0: input MLP + transformer block ---
  mlp_bn_run(stream, ws, x_in, mlp_input, X[0], Ns[0]);
  tblock_run(stream, ws, tb_input, X[0], POS[0], Ns[0], DIMS[0]);

  // --- encoder ---
  for (int i = 0; i < 4; ++i) {
    const int Ni = Ns[i], Nn = Ns[i + 1], Cn = DIMS[i + 1];
    size_t mark = ws.off;
    int* idc = ws.ip(Nn);
    float* dists = ws.f(Ni);
    fps_kernel<<<dim3(1), dim3(256), 0, stream>>>(POS[i], Ni, Nn, idc, dists);
    float* XM = ws.f((size_t)Ni * Cn);
    mlp_bn_run(stream, ws, X[i], enc_td[i], XM, Ni);
    gather_rows_kernel<<<blocks_for((size_t)Nn * 3), 256, 0, stream>>>(POS[i], idc, POSb[i + 1], Nn, 3);
    int* nb = ws.ip((size_t)Nn * 16);
    knn_kernel<<<blocks_for(Nn), 256, 0, stream>>>(POSb[i + 1], POS[i], Nn, Ni, 16, 0, nb);
    gather_max_kernel<<<blocks_for((size_t)Nn * Cn), 256, 0, stream>>>(XM, nb, X[i + 1], Nn, 16, Cn);
    ws.off = mark;
    tblock_run(stream, ws, enc_tb[i], X[i + 1], POS[i + 1], Nn, Cn);
  }

  // --- summit ---
  gemmL(stream, X[4], summit, XS, Ns[4], true);          // mlp_plain
  tblock_run(stream, ws, tb_summit, XS, POS[4], Ns[4], 512);

  // --- decoder ---
  float* cur = XS;                                       // level 4 features
  for (int i = 0; i < 4; ++i) {
    const int lvl = 3 - i;
    const int N = Ns[lvl], Cl = DIMS[lvl];
    const int Nsub = Ns[lvl + 1];
    size_t mark = ws.off;
    float* XSUB = ws.f((size_t)Nsub * Cl);
    mlp_bn_run(stream, ws, cur, dec_sub[i], XSUB, Nsub);
    int* idx3 = ws.ip((size_t)N * 3);
    knn_kernel<<<blocks_for(N), 256, 0, stream>>>(POS[lvl], POS[lvl + 1], N, Nsub, 3, 0, idx3);
    float* XINT = ws.f((size_t)N * Cl);
    interp3_kernel<<<blocks_for((size_t)N * Cl), 256, 0, stream>>>(XSUB, idx3, POS[lvl], POS[lvl + 1], XINT, N, Cl);
    mlp_bn_run(stream, ws, X[lvl], dec_mlp[i], X[lvl], N);       // in-place via tmp
    add_inplace_kernel<<<blocks_for((size_t)N * Cl), 256, 0, stream>>>(X[lvl], XINT, N * Cl);
    ws.off = mark;
    tblock_run(stream, ws, dec_tb[i], X[lvl], POS[lvl], N, Cl);
    cur = X[lvl];
  }

  // --- output head + log_softmax ---
  {
    size_t mark = ws.off;
    float* h1 = ws.f((size_t)Ns[0] * 64);
    float* h2 = ws.f((size_t)Ns[0] * 64);
    float* h3 = ws.f((size_t)Ns[0] * 13);
    gemmL(stream, cur, out0, h1, Ns[0], true);
    gemmL(stream, h1, out1, h2, Ns[0], true);
    gemmL(stream, h2, out2, h3, Ns[0], false);
    log_softmax_kernel<<<blocks_for(Ns[0]), 256, 0, stream>>>(h3, (float*)d_out, Ns[0], 13);
    ws.off = mark;
  }
}